// CategoricalNet_19507741459020
// MI455X (gfx1250) — compile-verified
//
#include <hip/hip_runtime.h>
#include <math.h>

// ---------------------------------------------------------------------------
// Top-k (k=50) + top-p (0.9) filtered softmax over rows of [2048, 50257] f32.
// Memory-bound: floor = (412MB in + 412MB out) / 23.3TB/s ~= 35us on MI455X.
// One 1024-thread (32-wave, wave32) block per row.
//
// Strategy: the top-50 of a 50257-wide row lives in a sparse tail, so a single
// 12-bit (sign+exp+3 mantissa bits) LDS histogram pass is enough to find a
// coarse bin B with suffix-count >= 50. Gather the ~60-300 elements with
// key >= B<<20 into LDS, finish exact top-k (incl. ties) + stable top-p cutoff
// with a tiny in-LDS rank sort, then one output pass. Only 3 full-row scans:
// the first is the single HBM read (row then stays L2-hot, 201KB vs 192MB L2),
// the write is the single HBM write -> kernel runs at the bandwidth roofline.
// ---------------------------------------------------------------------------

constexpr int   VOCAB       = 50257;
constexpr int   NT          = 1024;   // 32 wave32 waves per block
constexpr int   TOPK        = 50;
constexpr float TEMPERATURE = 0.8f;
constexpr float TOP_P       = 0.9f;

constexpr int NBIN  = 4096;           // 12-bit bins: sign + exp8 + mant3
constexpr int NCOPY = 2;              // replicated histograms (atomic relief)
constexpr int HPAD  = NBIN + 1;       // pad -> copies land on different banks
constexpr int CAP   = 1024;           // candidate buffer (expected ~60-300)

// Monotonic unsigned key: larger float  <=>  larger key.
__device__ __forceinline__ unsigned fkey(float f) {
    unsigned u = __float_as_uint(f);
    return u ^ (unsigned)(((int)u >> 31) | (int)0x80000000);
}

__global__ __launch_bounds__(NT) void topk_topp_softmax_kernel(
    const float* __restrict__ logits, float* __restrict__ out)
{
    __shared__ unsigned s_hist[NCOPY * HPAD];   // 32.8 KB
    __shared__ unsigned s_scan[NT];             //  4   KB
    __shared__ float    s_cval[CAP];            //  4   KB
    __shared__ int      s_cidx[CAP];            //  4   KB
    __shared__ float    s_sv[CAP];              //  4   KB
    __shared__ int      s_si[CAP];              //  4   KB
    __shared__ int      s_cnt, s_B, s_J;
    __shared__ float    s_m, s_invZ;
    __shared__ unsigned s_kcut;
    __shared__ int      s_icut;

    const int tid = threadIdx.x;
    const float* __restrict__ row  = logits + (size_t)blockIdx.x * VOCAB;
    float* __restrict__       orow = out    + (size_t)blockIdx.x * VOCAB;

    if (tid == 0) s_cnt = 0;

    // Warm the row toward L2 (gfx1250 global_prefetch_b8); one per 128B line.
    for (int i = tid * 32; i < VOCAB; i += NT * 32)
        __builtin_prefetch(row + i, 0, 3);

    // ---- pass 1: 12-bit histogram of the monotonic keys -------------------
    for (int i = tid; i < NCOPY * HPAD; i += NT) s_hist[i] = 0u;
    __syncthreads();

    unsigned* hrow = &s_hist[(tid & (NCOPY - 1)) * HPAD];
    for (int i = tid; i < VOCAB; i += NT)
        atomicAdd(&hrow[fkey(row[i]) >> 20], 1u);
    __syncthreads();

    // per-thread partial over 4 consecutive bins (both copies)
    {
        const int base = 4 * tid;
        unsigned p = 0;
#pragma unroll
        for (int k = 0; k < 4; ++k)
            p += s_hist[base + k] + s_hist[HPAD + base + k];
        s_scan[tid] = p;
    }
    __syncthreads();

    // inclusive suffix sum over the 1024 partials (count of keys >= chunk)
    for (int d = 1; d < NT; d <<= 1) {
        unsigned add = (tid + d < NT) ? s_scan[tid + d] : 0u;
        __syncthreads();
        s_scan[tid] += add;
        __syncthreads();
    }

    // unique crossing chunk: suffix[t] >= TOPK > suffix[t+1]; walk its 4 bins
    {
        unsigned sge  = s_scan[tid];
        unsigned sgeN = (tid == NT - 1) ? 0u : s_scan[tid + 1];
        if ((int)sge >= TOPK && (int)sgeN < TOPK) {
            unsigned above = sgeN;
            for (int b = 4 * tid + 3; b >= 4 * tid; --b) {
                unsigned c = s_hist[b] + s_hist[HPAD + b];
                if ((int)(above + c) >= TOPK) { s_B = b; break; }
                above += c;
            }
        }
    }
    __syncthreads();
    const unsigned Tg = (unsigned)s_B << 20;   // gather threshold (superset)

    // ---- pass 2: gather candidates (all of the true top-k are included) ----
    for (int i = tid; i < VOCAB; i += NT) {
        float x = row[i];
        if (fkey(x) >= Tg) {
            int p = atomicAdd(&s_cnt, 1);
            if (p < CAP) { s_cval[p] = x; s_cidx[p] = i; }
        }
    }
    __syncthreads();
    const int N = min(s_cnt, CAP);

    // ---- stable rank sort: (value desc, index asc) == jnp.argsort(-l) stable
    if (tid < N) {
        float v = s_cval[tid]; int id = s_cidx[tid];
        int r = 0;
        for (int j = 0; j < N; ++j) {
            float vj = s_cval[j]; int ij = s_cidx[j];
            r += (vj > v) || (vj == v && ij < id);
        }
        s_sv[r] = v; s_si[r] = id;
    }
    __syncthreads();

    // top-k survivors = sorted[0..J-1]: everything >= the 50th value (ties in)
    const float vk = s_sv[TOPK - 1];
    if (tid == 0) {
        int j = TOPK;
        while (j < N && s_sv[j] == vk) ++j;
        s_J = j;
        s_m = s_sv[0] / TEMPERATURE;           // row max of scaled logits
    }
    __syncthreads();
    const int   J = s_J;
    const float m = s_m;

    // exp(l - m) per survivor (parallel); reuse s_cval as the e[] buffer
    if (tid < J)
        s_cval[tid] = expf(s_sv[tid] / TEMPERATURE - m);
    __syncthreads();

    // ---- sequential cumsum (matches reference cumsum order) to find cutoff
    if (tid == 0) {
        float tot = 0.0f;
        for (int j = 0; j < J; ++j) tot += s_cval[j];
        const float inv_tot = 1.0f / tot;

        float cum = 0.0f, zkeep = 0.0f;
        int kstar = J - 1;
        for (int j = 0; j < J; ++j) {
            zkeep += s_cval[j];
            cum   += s_cval[j] * inv_tot;
            if (cum > TOP_P) { kstar = j; break; }  // first token over P kept
        }
        s_invZ = 1.0f / zkeep;                 // renorm over the kept prefix
        s_kcut = fkey(s_sv[kstar]);            // cutoff value (as key) ...
        s_icut = s_si[kstar];                  // ... with stable index tiebreak
    }
    __syncthreads();

    // ---- pass 3: write probs. exp is branch-skipped: kept tokens are ~50 of
    // 50257, so nearly every wave takes the execz path and this is pure stream.
    const float    invZ = s_invZ;
    const unsigned kcut = s_kcut;
    const int      icut = s_icut;
    for (int i = tid; i < VOCAB; i += NT) {
        float    x = row[i];
        unsigned k = fkey(x);
        float    p = 0.0f;
        if (k > kcut || (k == kcut && i <= icut))
            p = expf(x / TEMPERATURE - m) * invZ;
        orow[i] = p;
    }
}

extern "C" void kernel_launch(void* const* d_in, const int* in_sizes, int n_in,
                              void* d_out, int out_size, void* d_ws, size_t ws_size,
                              hipStream_t stream) {
    (void)n_in; (void)out_size; (void)d_ws; (void)ws_size;
    const float* logits = (const float*)d_in[0];
    float*       out    = (float*)d_out;
    const int batch = in_sizes[0] / VOCAB;   // 2048 rows
    if (batch <= 0) return;
    topk_topp_softmax_kernel<<<batch, NT, 0, stream>>>(logits, out);
}